// RuleConstraintLoss_31086973288940
// MI455X (gfx1250) — compile-verified
//
#include <hip/hip_runtime.h>
#include <math.h>

// ---------------- problem constants (match reference) ----------------
constexpr int B = 32, L = 4096, D = 512, M = 256;
constexpr int OP_IDENTITY = 2, OP_CANCEL_START = 3, OP_CANCEL_END = 4, OP_STAR_ZERO = 5;
constexpr float IDENTITY_W = 1.0f, CANCEL_W = 1.0f, COLLAPSE_W = 0.5f;
constexpr float EPSF = 1e-9f;

constexpr int WAVES_PER_BLOCK = 8;            // 256 threads = 8 x wave32 (gfx1250 is wave32-only)
constexpr int NOPS = B * M;                   // 8192 ops, one wave each
constexpr int NBLK = NOPS / WAVES_PER_BLOCK;  // 1024 blocks
constexpr int FPL = D / 32;                   // 16 floats per lane

typedef __attribute__((ext_vector_type(4))) float v4f;

// gfx1250 async global->LDS path (ASYNCcnt), guarded so compile never breaks.
#if __has_builtin(__builtin_amdgcn_global_load_async_to_lds_b128) && \
    __has_builtin(__builtin_amdgcn_s_wait_asynccnt)
#define USE_ASYNC_LDS 1
// Builtin signature (from clang diagnostic): param0 = int __vector_size__(16) __device__*  (AS1)
typedef int v4i_ __attribute__((vector_size(16)));
typedef __attribute__((address_space(1))) v4i_ gv4i;  // global (printed as "__device__")
typedef __attribute__((address_space(3))) v4i_ lv4i;  // LDS    (printed as "__shared__")
#else
#define USE_ASYNC_LDS 0
#endif

__device__ __forceinline__ float wave_sum(float v) {
#pragma unroll
  for (int off = 16; off >= 1; off >>= 1) v += __shfl_xor(v, off, 32);
  return v;  // all 32 lanes hold the total
}

__device__ __forceinline__ void load_row_direct(const float* __restrict__ row,
                                                int lane, float v[FPL]) {
  const v4f* p = (const v4f*)row + lane * (FPL / 4);
#pragma unroll
  for (int i = 0; i < FPL / 4; ++i) {
    v4f t = __builtin_nontemporal_load(p + i);  // NT: 256MB tensor > 192MB L2, rows ~single-use
    v[4 * i + 0] = t.x; v[4 * i + 1] = t.y; v[4 * i + 2] = t.z; v[4 * i + 3] = t.w;
  }
}

#if USE_ASYNC_LDS
__device__ __forceinline__ void async_stage_row(const float* __restrict__ row,
                                                float* __restrict__ lbuf, int lane) {
  // Each lane moves one 16B chunk per issue; 4 issues cover the 2KB row.
#pragma unroll
  for (int i = 0; i < 4; ++i) {
    const float* g = row + i * 128 + lane * 4;
    float* l = lbuf + i * 128 + lane * 4;
    __builtin_amdgcn_global_load_async_to_lds_b128((gv4i*)g, (lv4i*)l, 0, 0);
  }
}

__device__ __forceinline__ void lds_read_row(const float* __restrict__ lbuf,
                                             int lane, float v[FPL]) {
  const v4f* p = (const v4f*)lbuf + lane * (FPL / 4);
#pragma unroll
  for (int i = 0; i < FPL / 4; ++i) {
    v4f t = p[i];
    v[4 * i + 0] = t.x; v[4 * i + 1] = t.y; v[4 * i + 2] = t.z; v[4 * i + 3] = t.w;
  }
}
#endif

__global__ __launch_bounds__(256) void rule_loss_kernel(
    const float* __restrict__ states, const int* __restrict__ op_types,
    const int* __restrict__ op_bp, const int* __restrict__ op_ap,
    const int* __restrict__ n_ops, float* __restrict__ partials) {
#if USE_ASYNC_LDS
  __shared__ alignas(16) float stage[WAVES_PER_BLOCK][2][D];  // 32KB of 320KB/WGP LDS
#endif
  __shared__ float acc[WAVES_PER_BLOCK][6];

  const int wave = threadIdx.x >> 5;
  const int lane = threadIdx.x & 31;
  const int o = blockIdx.x * WAVES_PER_BLOCK + wave;  // op id
  const int b = o / M;
  const int m = o % M;

  const int n  = n_ops[b];
  const int ty = op_types[b * M + m];
  const int bp = op_bp[b * M + m];
  const int ap = op_ap[b * M + m];

  const bool valid = (m < n) & (bp >= 0) & (bp < L) & (ap >= 0) & (ap < L);

  float vals[3] = {0.f, 0.f, 0.f};  // identity, cancel, collapse contributions
  float cnts[3] = {0.f, 0.f, 0.f};

  const float* bbase = states + (size_t)b * L * D;

  // Decide branch + the (at most two) rows this op needs. Wave-uniform.
  int rowA = 0, rowB = 0, branch = -1;  // A = before, B = after / after_pair
  if (valid) {
    if (ty == OP_IDENTITY) { branch = 0; rowA = bp; rowB = ap; }
    else if (ty == OP_CANCEL_START) {
      const int tyn = (m + 1 < M) ? op_types[b * M + m + 1] : -1;
      const int apn = (m + 1 < M) ? op_ap[b * M + m + 1] : L;
      if (((m + 1) < n) && (tyn == OP_CANCEL_END) && (apn < L)) {
        branch = 1; rowA = bp; rowB = min(max(apn, 0), L - 1);
      }
    } else if (ty == OP_STAR_ZERO) { branch = 2; rowA = bp; rowB = ap; }
  }

  if (branch >= 0) {
    const float* rA = bbase + (size_t)rowA * D;
    const float* rB = bbase + (size_t)rowB * D;
    __builtin_prefetch(rA + lane * FPL, 0, 0);  // global_prefetch_b8
    __builtin_prefetch(rB + lane * FPL, 0, 0);

    float va[FPL], vb[FPL];
#if USE_ASYNC_LDS
    async_stage_row(rA, &stage[wave][0][0], lane);   // both rows in flight together
    async_stage_row(rB, &stage[wave][1][0], lane);
    __builtin_amdgcn_s_wait_asynccnt(0);             // s_wait_asynccnt 0
    lds_read_row(&stage[wave][0][0], lane, va);
    lds_read_row(&stage[wave][1][0], lane, vb);
#else
    load_row_direct(rA, lane, va);
    load_row_direct(rB, lane, vb);
#endif

    if (branch == 2) {
      // energy-normalized entropy hinge: relu(H(after) - H(before) + 0.5)
      float eb = 0.f, ea = 0.f;
#pragma unroll
      for (int i = 0; i < FPL; ++i) { eb += va[i] * va[i]; ea += vb[i] * vb[i]; }
      const float Sb = wave_sum(eb) + EPSF;
      const float Sa = wave_sum(ea) + EPSF;
      float hb = 0.f, ha = 0.f;
#pragma unroll
      for (int i = 0; i < FPL; ++i) {
        const float pb = (va[i] * va[i]) / Sb;
        const float pa = (vb[i] * vb[i]) / Sa;
        hb -= pb * log2f(pb + EPSF);
        ha -= pa * log2f(pa + EPSF);
      }
      const float t = wave_sum(ha) - wave_sum(hb) + 0.5f;
      vals[2] = t > 0.f ? t : 0.f;
      cnts[2] = 1.f;
    } else {
      float s = 0.f;
#pragma unroll
      for (int i = 0; i < FPL; ++i) { const float d = vb[i] - va[i]; s += d * d; }
      vals[branch] = wave_sum(s) * (1.0f / (float)D);  // mean over D
      cnts[branch] = 1.f;
    }
  }

  if (lane == 0) {
    acc[wave][0] = vals[0]; acc[wave][1] = vals[1]; acc[wave][2] = vals[2];
    acc[wave][3] = cnts[0]; acc[wave][4] = cnts[1]; acc[wave][5] = cnts[2];
  }
  __syncthreads();
  if (threadIdx.x < 6) {
    float s = 0.f;
#pragma unroll
    for (int w = 0; w < WAVES_PER_BLOCK; ++w) s += acc[w][threadIdx.x];
    partials[blockIdx.x * 6 + threadIdx.x] = s;  // fully overwritten each call
  }
}

__global__ __launch_bounds__(256) void finalize_kernel(
    const float* __restrict__ partials, float* __restrict__ out) {
  __shared__ float red[256][6];
  float a[6] = {0.f, 0.f, 0.f, 0.f, 0.f, 0.f};
  for (int i = threadIdx.x; i < NBLK; i += 256)
#pragma unroll
    for (int j = 0; j < 6; ++j) a[j] += partials[i * 6 + j];
#pragma unroll
  for (int j = 0; j < 6; ++j) red[threadIdx.x][j] = a[j];
  __syncthreads();
  for (int s = 128; s >= 1; s >>= 1) {
    if ((int)threadIdx.x < s)
#pragma unroll
      for (int j = 0; j < 6; ++j) red[threadIdx.x][j] += red[threadIdx.x + s][j];
    __syncthreads();
  }
  if (threadIdx.x == 0) {
    const float idl = red[0][0] / fmaxf(red[0][3], 1.f);
    const float cal = red[0][1] / fmaxf(red[0][4], 1.f);
    const float col = red[0][2] / fmaxf(red[0][5], 1.f);
    out[0] = IDENTITY_W * idl + CANCEL_W * cal + COLLAPSE_W * col;
    out[1] = idl;
    out[2] = cal;
    out[3] = col;
  }
}

extern "C" void kernel_launch(void* const* d_in, const int* in_sizes, int n_in,
                              void* d_out, int out_size, void* d_ws, size_t ws_size,
                              hipStream_t stream) {
  (void)in_sizes; (void)n_in; (void)out_size; (void)ws_size;
  const float* states  = (const float*)d_in[0];
  const int* op_types  = (const int*)d_in[1];
  const int* op_bp     = (const int*)d_in[2];
  const int* op_ap     = (const int*)d_in[3];
  const int* n_ops     = (const int*)d_in[4];
  float* out      = (float*)d_out;
  float* partials = (float*)d_ws;   // NBLK * 6 floats = 24KB

  rule_loss_kernel<<<NBLK, 256, 0, stream>>>(states, op_types, op_bp, op_ap, n_ops, partials);
  finalize_kernel<<<1, 256, 0, stream>>>(partials, out);
}